// AlphaRotatedIoULoss_81956565942365
// MI455X (gfx1250) — compile-verified
//
#include <hip/hip_runtime.h>
#include <hip/hip_bf16.h>
#include <math.h>

#define BLOCK 256
#define MAXBLOCKS 2048
#define EPS 1e-6f

typedef __attribute__((ext_vector_type(2))) float v2f;
typedef __attribute__((ext_vector_type(8))) float v8f;

#if defined(__has_builtin)
#if __has_builtin(__builtin_amdgcn_wmma_f32_16x16x4_f32)
#define USE_WMMA_REDUCE 1
#endif
#endif

// Wave32 sum of one float per lane. Uses V_WMMA_F32_16X16X4_F32 with an
// all-ones A matrix: D[0][n] = sum_k B[k][n], i.e. row 0 of D (lanes 0-15 of
// d[0], mirrored in lanes 16-31) holds 16 column sums whose total equals the
// sum over every B storage slot = sum over all 32 lanes. Four xor-shuffles
// finish the reduction. Exact f32 arithmetic. EXEC must be all-1s at the
// call site (callers keep all 256 threads converged here).
__device__ __forceinline__ float wave_sum(float v) {
#if USE_WMMA_REDUCE
  v2f a; a[0] = 1.0f; a[1] = 1.0f;       // A = ones (16x4 f32, 2 VGPRs/lane)
  v2f b; b[0] = v;    b[1] = 0.0f;       // per-lane partial into B
  v8f c = {};
  v8f d = __builtin_amdgcn_wmma_f32_16x16x4_f32(
      /*neg_a=*/false, a, /*neg_b=*/false, b,
      /*c_mod=*/(short)0, c, /*reuse_a=*/false, /*reuse_b=*/false);
  float s = d[0];
  s += __shfl_xor(s, 8, 32);
  s += __shfl_xor(s, 4, 32);
  s += __shfl_xor(s, 2, 32);
  s += __shfl_xor(s, 1, 32);
  return s;
#else
  for (int off = 16; off > 0; off >>= 1) v += __shfl_xor(v, off, 32);
  return v;
#endif
}

// Per-block partial sums of the alpha-rotated-IoU loss.
// Intersection of the two rotated rectangles via Sutherland-Hodgman clipping
// (convex-convex, <= 8 vertices). Polygon scratch lives in LDS, padded to
// 9 float2 slots/thread so wave32 lanes hit disjoint bank pairs.
__global__ void __launch_bounds__(BLOCK)
riou_loss_partials(const float* __restrict__ pred,
                   const float* __restrict__ target,
                   const float* __restrict__ weight,
                   float* __restrict__ partial, int n) {
  __shared__ float2 bufA[BLOCK * 9];
  __shared__ float2 bufB[BLOCK * 9];
  __shared__ float wsum[BLOCK / 32];

  const int stride = gridDim.x * BLOCK;
  float lsum = 0.0f;

  for (int i = blockIdx.x * BLOCK + threadIdx.x; i < n; i += stride) {
    if (i + stride < n) {  // gfx1250: global_prefetch_b8 for the next tile
      __builtin_prefetch(pred + (size_t)(i + stride) * 5, 0, 1);
      __builtin_prefetch(target + (size_t)(i + stride) * 5, 0, 1);
    }
    const float* bp = pred + (size_t)i * 5;
    const float* bt = target + (size_t)i * 5;
    float p_x = bp[0], p_y = bp[1], p_w = bp[2], p_h = bp[3], p_a = bp[4];
    float t_x = bt[0], t_y = bt[1], t_w = bt[2], t_h = bt[3], t_a = bt[4];

    float s1, co1; sincosf(p_a, &s1, &co1);
    float s2, co2; sincosf(t_a, &s2, &co2);

    // Subject polygon (pred corners, CCW) into LDS.
    float2* P = &bufA[threadIdx.x * 9];
    float2* Q = &bufB[threadIdx.x * 9];
    {
      float hx = 0.5f * p_w, hy = 0.5f * p_h;
      P[0] = make_float2(p_x + hx * co1 - hy * s1, p_y + hx * s1 + hy * co1);
      P[1] = make_float2(p_x - hx * co1 - hy * s1, p_y - hx * s1 + hy * co1);
      P[2] = make_float2(p_x - hx * co1 + hy * s1, p_y - hx * s1 - hy * co1);
      P[3] = make_float2(p_x + hx * co1 + hy * s1, p_y + hx * s1 - hy * co1);
    }
    // Clip polygon (target corners, CCW) kept in registers (constant indexed
    // after full unroll of the edge loop).
    float c2x[4], c2y[4];
    {
      float hx = 0.5f * t_w, hy = 0.5f * t_h;
      c2x[0] = t_x + hx * co2 - hy * s2; c2y[0] = t_y + hx * s2 + hy * co2;
      c2x[1] = t_x - hx * co2 - hy * s2; c2y[1] = t_y - hx * s2 + hy * co2;
      c2x[2] = t_x - hx * co2 + hy * s2; c2y[2] = t_y - hx * s2 - hy * co2;
      c2x[3] = t_x + hx * co2 + hy * s2; c2y[3] = t_y + hx * s2 - hy * co2;
    }

    int cnt = 4;
#pragma unroll
    for (int e = 0; e < 4; ++e) {
      float ax = c2x[e], ay = c2y[e];
      float ex = c2x[(e + 1) & 3] - ax;
      float ey = c2y[(e + 1) & 3] - ay;
      int m = 0;
      for (int v = 0; v < cnt; ++v) {
        float2 cur = P[v];
        float2 nxt = P[(v + 1 == cnt) ? 0 : v + 1];
        float di = ex * (cur.y - ay) - ey * (cur.x - ax);  // >=0 : inside (left of CCW edge)
        float dj = ex * (nxt.y - ay) - ey * (nxt.x - ax);
        bool in_i = di >= 0.0f;
        bool in_j = dj >= 0.0f;
        if (in_i) Q[m++] = cur;
        if (in_i != in_j) {
          float t = di / (di - dj);
          Q[m++] = make_float2(cur.x + t * (nxt.x - cur.x),
                               cur.y + t * (nxt.y - cur.y));
        }
      }
      cnt = m;
      float2* tmp = P; P = Q; Q = tmp;
    }

    float area2 = 0.0f;  // shoelace (x2)
    for (int v = 0; v < cnt; ++v) {
      float2 cur = P[v];
      float2 nxt = P[(v + 1 == cnt) ? 0 : v + 1];
      area2 += cur.x * nxt.y - cur.y * nxt.x;
    }
    float inter = 0.5f * fabsf(area2);

    float uni = fmaxf(p_w * p_h + t_w * t_h - inter, EPS);
    float iou = fmaxf(inter / uni, EPS);
    float loss = 1.0f - iou * iou * iou;  // ALPHA = 3
    lsum += loss * weight[i];
  }

  // All 256 threads converged here: EXEC all-1s for the WMMA reduction.
  float wtot = wave_sum(lsum);
  int lane = threadIdx.x & 31;
  int wid = threadIdx.x >> 5;
  if (lane == 0) wsum[wid] = wtot;
  __syncthreads();
  if (threadIdx.x == 0) {
    float t = 0.0f;
#pragma unroll
    for (int w = 0; w < BLOCK / 32; ++w) t += wsum[w];
    partial[blockIdx.x] = t;
  }
}

__global__ void __launch_bounds__(BLOCK)
riou_loss_finalize(const float* __restrict__ partial, int nb,
                   float* __restrict__ out, float inv_n) {
  __shared__ float wsum[BLOCK / 32];
  float s = 0.0f;
  for (int i = threadIdx.x; i < nb; i += BLOCK) s += partial[i];
  float wtot = wave_sum(s);
  int lane = threadIdx.x & 31;
  int wid = threadIdx.x >> 5;
  if (lane == 0) wsum[wid] = wtot;
  __syncthreads();
  if (threadIdx.x == 0) {
    float t = 0.0f;
#pragma unroll
    for (int w = 0; w < BLOCK / 32; ++w) t += wsum[w];
    out[0] = t * inv_n;  // LOSS_WEIGHT == 1.0
  }
}

extern "C" void kernel_launch(void* const* d_in, const int* in_sizes, int n_in,
                              void* d_out, int out_size, void* d_ws, size_t ws_size,
                              hipStream_t stream) {
  const float* pred = (const float*)d_in[0];
  const float* target = (const float*)d_in[1];
  const float* weight = (const float*)d_in[2];
  float* out = (float*)d_out;
  float* partial = (float*)d_ws;

  int n = in_sizes[0] / 5;
  int nb = (n + BLOCK - 1) / BLOCK;
  if (nb > MAXBLOCKS) nb = MAXBLOCKS;
  if (nb < 1) nb = 1;

  riou_loss_partials<<<nb, BLOCK, 0, stream>>>(pred, target, weight, partial, n);
  riou_loss_finalize<<<1, BLOCK, 0, stream>>>(partial, nb, out,
                                              n > 0 ? 1.0f / (float)n : 0.0f);
}